// ConsciousMoE_41403484733947
// MI455X (gfx1250) — compile-verified
//
#include <hip/hip_runtime.h>
#include <hip/hip_bf16.h>
#include <math.h>

typedef __attribute__((ext_vector_type(2))) float v2f;
typedef __attribute__((ext_vector_type(8))) float v8f;

#define HIDDEN   2048
#define EXP_HID  4096
#define VOCAB    4096
#define NEXP     16
#define NCELLS   64
#define TOPK     2

#define KC          32    // K-rows per staged chunk
#define LDS_STRIDE  144   // dwords per staged row: 128 + 16 pad (conflict-free row+2 reads)

// ---- CDNA5 async global->LDS path (guarded; falls back to sync copy) ------
#if defined(__AMDGCN__) && __has_builtin(__builtin_amdgcn_global_load_async_to_lds_b128)
#define HAVE_ASYNC_LDS 1
#else
#define HAVE_ASYNC_LDS 0
#endif

#if HAVE_ASYNC_LDS
#if __has_builtin(__builtin_amdgcn_s_wait_asynccnt)
#define WAIT_ASYNC(n) __builtin_amdgcn_s_wait_asynccnt(n)
#else
#define WAIT_ASYNC(n) asm volatile("s_wait_asynccnt %0" ::"i"(n) : "memory")
#endif
typedef int v16b __attribute__((vector_size(4 * sizeof(int))));
typedef __attribute__((address_space(1))) v16b* as1_v16b;
typedef __attribute__((address_space(3))) v16b* as3_v16b;
#else
#define WAIT_ASYNC(n) ((void)0)
#endif

__device__ __forceinline__ void stage_b128(const float* __restrict__ g, float* l) {
#if HAVE_ASYNC_LDS
  __builtin_amdgcn_global_load_async_to_lds_b128((as1_v16b)g, (as3_v16b)l, 0, 0);
#else
  *(float4*)l = *(const float4*)g;   // coalesced 128B/line fallback
#endif
}

// ---------------------------------------------------------------------------
// Kernel 1: c = mean(states, axis=0); router logits; softmax top-2 (1 block)
// ---------------------------------------------------------------------------
__global__ __launch_bounds__(256)
void moe_router_kernel(const float* __restrict__ states,
                       const float* __restrict__ Wr,
                       const float* __restrict__ br,
                       float* __restrict__ c_out,
                       float* __restrict__ vals_out,
                       int* __restrict__ idx_out) {
  __shared__ float c_lds[HIDDEN];
  __shared__ float logits[NEXP];
  const int t = threadIdx.x;

  for (int j = t; j < HIDDEN; j += 256) {
    float s = 0.f;
#pragma unroll 8
    for (int i = 0; i < NCELLS; ++i) s += states[i * HIDDEN + j];
    const float m = s * (1.0f / NCELLS);
    c_lds[j] = m;
    c_out[j] = m;
  }
  __syncthreads();

  if (t < NEXP) {
    float acc = br[t];
    for (int j = 0; j < HIDDEN; ++j) acc += c_lds[j] * Wr[j * NEXP + t];
    logits[t] = acc;
  }
  __syncthreads();

  if (t == 0) {
    int i0 = 0; float v0 = logits[0];
    for (int e = 1; e < NEXP; ++e)
      if (logits[e] > v0) { v0 = logits[e]; i0 = e; }
    int i1 = -1; float v1 = -INFINITY;
    for (int e = 0; e < NEXP; ++e)
      if (e != i0 && logits[e] > v1) { v1 = logits[e]; i1 = e; }
    // softmax normalization cancels in vals/vals.sum(): only top-2 logits matter
    const float e1 = expf(v1 - v0);
    const float inv = 1.0f / (1.0f + e1);
    vals_out[0] = inv;
    vals_out[1] = e1 * inv;
    idx_out[0] = i0;
    idx_out[1] = i1;
  }
}

// ---------------------------------------------------------------------------
// Kernel 2: up-proj GEMV, split-K=2. Block = 8 waves = 128 contiguous cols.
// Double-buffered async global->LDS staging of 32-row weight panels (full
// 128B-line b128 transfers), WMMA f32 16x16x4 fed from LDS. Raw partials out.
// ---------------------------------------------------------------------------
__global__ __launch_bounds__(256)
void moe_up_kernel(const float* __restrict__ Wup,    // [E, HIDDEN, EXP_HID]
                   const float* __restrict__ c_in,   // [HIDDEN]
                   const int* __restrict__ idx_in,   // [2]
                   float* __restrict__ hpart)        // [2 split][2 k][EXP_HID]
{
  __shared__ float buf[2][KC * LDS_STRIDE];
  __shared__ float c_lds[HIDDEN / 2];

  const int t      = threadIdx.x;
  const int k      = blockIdx.x >> 6;          // expert slot 0/1
  const int split  = (blockIdx.x >> 5) & 1;    // K split 0/1
  const int krows  = HIDDEN / 2;               // 1024
  const int kbase  = split * krows;
  const int colblk = (blockIdx.x & 31) * 128;

  for (int j = t; j < krows; j += 256) c_lds[j] = c_in[kbase + j];

  const int lane  = t & 31;
  const int wave  = t >> 5;
  const int half  = lane >> 4;
  const int n     = lane & 15;
  const int lane4 = lane * 4;

  const int e = idx_in[k];
  const float* W = Wup + (size_t)e * HIDDEN * EXP_HID
                       + (size_t)kbase * EXP_HID + colblk;

  // prefetch chunk 0
#pragma unroll
  for (int p = 0; p < 4; ++p) {
    const int rr = p * 8 + wave;
    stage_b128(W + (size_t)rr * EXP_HID + lane4, &buf[0][rr * LDS_STRIDE + lane4]);
  }

  v8f acc = {};
  const int nchunks = krows / KC;              // 32
  for (int i = 0; i < nchunks; ++i) {
    if (i + 1 < nchunks) {
      const float* src = W + (size_t)(i + 1) * KC * EXP_HID;
      float* dst = &buf[(i + 1) & 1][0];
#pragma unroll
      for (int p = 0; p < 4; ++p) {
        const int rr = p * 8 + wave;
        stage_b128(src + (size_t)rr * EXP_HID + lane4, dst + rr * LDS_STRIDE + lane4);
      }
      WAIT_ASYNC(4);   // in-order: oldest 4 (chunk i) complete
    } else {
      WAIT_ASYNC(0);
    }
    __syncthreads();

    const float* bb = &buf[i & 1][(2 * half) * LDS_STRIDE + wave * 16 + n];
    const float* aa = &c_lds[i * KC + 2 * half];
#pragma unroll
    for (int d = 0; d < KC; d += 4) {
      v2f a, b;
      a.x = aa[d];
      a.y = aa[d + 1];
      b.x = bb[d * LDS_STRIDE];
      b.y = bb[(d + 1) * LDS_STRIDE];
      acc = __builtin_amdgcn_wmma_f32_16x16x4_f32(
          false, a, false, b, (short)0, acc, false, false);
    }
    __syncthreads();
  }

  if (lane < 16)  // D row M=0 in element 0, lanes 0..15 (N = lane)
    hpart[(size_t)(split * 2 + k) * EXP_HID + colblk + wave * 16 + n] = acc[0];
}

// ---------------------------------------------------------------------------
// Kernel 3: down-proj GEMV, split-K=4, same staged pipeline. h-slice =
// gelu(p0+p1+bup) computed into LDS on the fly; vals[k]-weighted partials.
// ---------------------------------------------------------------------------
__global__ __launch_bounds__(256)
void moe_down_kernel(const float* __restrict__ Wdown,  // [E, EXP_HID, VOCAB]
                     const float* __restrict__ bup,    // [E, EXP_HID]
                     const float* __restrict__ bdown,  // [E, VOCAB]
                     const float* __restrict__ hpart,  // [2][2][EXP_HID]
                     const float* __restrict__ vals_in,
                     const int* __restrict__ idx_in,
                     float* __restrict__ dpart)        // [4 split][VOCAB]
{
  __shared__ float buf[2][KC * LDS_STRIDE];
  __shared__ float h_lds[EXP_HID / 4];

  const int t      = threadIdx.x;
  const int split  = blockIdx.x >> 5;          // 0..3
  const int krows  = EXP_HID / 4;              // 1024
  const int kbase  = split * krows;
  const int colblk = (blockIdx.x & 31) * 128;

  const int lane  = t & 31;
  const int wave  = t >> 5;
  const int half  = lane >> 4;
  const int n     = lane & 15;
  const int lane4 = lane * 4;

  float total = 0.f;
  for (int k = 0; k < TOPK; ++k) {
    const int e = idx_in[k];
    __syncthreads();   // protect h_lds rewrite vs previous expert's readers
    for (int j = t; j < krows; j += 256) {
      const int jj = kbase + j;
      const float x = hpart[(size_t)k * EXP_HID + jj]
                    + hpart[(size_t)(2 + k) * EXP_HID + jj]
                    + bup[(size_t)e * EXP_HID + jj];
      h_lds[j] = 0.5f * x * (1.0f + erff(x * 0.70710678118654752f));  // exact gelu
    }

    const float* W = Wdown + (size_t)e * EXP_HID * VOCAB
                           + (size_t)kbase * VOCAB + colblk;
#pragma unroll
    for (int p = 0; p < 4; ++p) {
      const int rr = p * 8 + wave;
      stage_b128(W + (size_t)rr * VOCAB + lane4, &buf[0][rr * LDS_STRIDE + lane4]);
    }

    v8f acc = {};
    const int nchunks = krows / KC;            // 32
    for (int i = 0; i < nchunks; ++i) {
      if (i + 1 < nchunks) {
        const float* src = W + (size_t)(i + 1) * KC * VOCAB;
        float* dst = &buf[(i + 1) & 1][0];
#pragma unroll
        for (int p = 0; p < 4; ++p) {
          const int rr = p * 8 + wave;
          stage_b128(src + (size_t)rr * VOCAB + lane4, dst + rr * LDS_STRIDE + lane4);
        }
        WAIT_ASYNC(4);
      } else {
        WAIT_ASYNC(0);
      }
      __syncthreads();

      const float* bb = &buf[i & 1][(2 * half) * LDS_STRIDE + wave * 16 + n];
      const float* aa = &h_lds[i * KC + 2 * half];
#pragma unroll
      for (int d = 0; d < KC; d += 4) {
        v2f a, b;
        a.x = aa[d];
        a.y = aa[d + 1];
        b.x = bb[d * LDS_STRIDE];
        b.y = bb[(d + 1) * LDS_STRIDE];
        acc = __builtin_amdgcn_wmma_f32_16x16x4_f32(
            false, a, false, b, (short)0, acc, false, false);
      }
      __syncthreads();
    }

    float r = acc[0];
    if (split == 0) r += bdown[(size_t)e * VOCAB + colblk + wave * 16 + n];
    total += vals_in[k] * r;
  }

  if (lane < 16)
    dpart[(size_t)split * VOCAB + colblk + wave * 16 + n] = total;
}

// ---------------------------------------------------------------------------
// Kernel 4: fixed-order split-K reduction (deterministic, no float atomics)
// ---------------------------------------------------------------------------
__global__ __launch_bounds__(256)
void moe_reduce_kernel(const float* __restrict__ dpart, float* __restrict__ out) {
  const int v = blockIdx.x * 256 + threadIdx.x;
  if (v < VOCAB)
    out[v] = ((dpart[v] + dpart[VOCAB + v])
            + (dpart[2 * VOCAB + v] + dpart[3 * VOCAB + v]));
}

extern "C" void kernel_launch(void* const* d_in, const int* in_sizes, int n_in,
                              void* d_out, int out_size, void* d_ws, size_t ws_size,
                              hipStream_t stream) {
  const float* states = (const float*)d_in[0];
  const float* Wr     = (const float*)d_in[1];
  const float* br     = (const float*)d_in[2];
  const float* Wup    = (const float*)d_in[3];
  const float* bup    = (const float*)d_in[4];
  const float* Wdown  = (const float*)d_in[5];
  const float* bdown  = (const float*)d_in[6];
  float* out = (float*)d_out;

  // workspace layout (floats)
  float* ws    = (float*)d_ws;
  float* c     = ws;                       // HIDDEN
  float* hpart = c + HIDDEN;               // 2*2*EXP_HID
  float* dpart = hpart + 4 * EXP_HID;      // 4*VOCAB
  float* vals  = dpart + 4 * VOCAB;        // 2
  int*   eidx  = (int*)(vals + 2);         // 2

  moe_router_kernel<<<1, 256, 0, stream>>>(states, Wr, br, c, vals, eidx);
  moe_up_kernel<<<128, 256, 0, stream>>>(Wup, c, eidx, hpart);
  moe_down_kernel<<<128, 256, 0, stream>>>(Wdown, bup, bdown, hpart, vals, eidx, dpart);
  moe_reduce_kernel<<<VOCAB / 256, 256, 0, stream>>>(dpart, out);
}